// GCNTopK_8521215115394
// MI455X (gfx1250) — compile-verified
//
#include <hip/hip_runtime.h>
#include <hip/hip_bf16.h>

typedef __attribute__((ext_vector_type(2))) float v2f;
typedef __attribute__((ext_vector_type(8))) float v8f;

#define HWF 128
#define EPSV 1e-5f

// ---------------------------------------------------------------------------
// Fused (optional BN->ReLU) -> Linear GEMM using V_WMMA_F32_16X16X4_F32.
// One wave computes a 16x16 tile of Y = act(X) @ W (+ bias).
// Block = 32 * (NOUT/16) threads; grid.x = n_rows/16 (exact multiple, EXEC all 1s).
// A fragment: lane L holds X[row = tile*16 + (L&15)][k0 + 2*(L>>4) + v], v=0,1
// B fragment: lane L holds W[k0 + 2*(L>>4) + v][colBase + (L&15)]
// C fragment: lane L holds rows { (L<16 ? r : r+8) } of column colBase + (L&15)
// ---------------------------------------------------------------------------
__global__ void lin16_wmma(const float* __restrict__ X, int K,
                           const float* __restrict__ W, int NOUT,
                           const float* __restrict__ gamma,
                           const float* __restrict__ beta,
                           const float* __restrict__ mean,
                           const float* __restrict__ var,
                           const float* __restrict__ bias,
                           float* __restrict__ Y, int use_act) {
  __shared__ float s_sc[HWF];
  __shared__ float s_sh[HWF];
  int t = threadIdx.x;
  if (use_act && t < K) {
    float sc = gamma[t] * rsqrtf(var[t] + EPSV);
    s_sc[t] = sc;
    s_sh[t] = beta[t] - mean[t] * sc;
  }
  __syncthreads();

  int lane = t & 31;
  int wv = t >> 5;                 // which 16-column tile
  int m = lane & 15;
  int kh = lane >> 4;              // 0 or 1: k-half within the 4-wide K step
  long row = (long)blockIdx.x * 16 + m;
  const float* xr = X + row * K;
  int colBase = wv * 16;
  int col = colBase + m;           // for B/C fragments n = lane & 15 == m

  v8f acc = {};
  for (int k0 = 0; k0 < K; k0 += 4) {
    int ka = k0 + 2 * kh;
    float a0 = xr[ka];
    float a1 = xr[ka + 1];
    if (use_act) {
      a0 = fmaxf(fmaf(a0, s_sc[ka],     s_sh[ka]),     0.0f);
      a1 = fmaxf(fmaf(a1, s_sc[ka + 1], s_sh[ka + 1]), 0.0f);
    }
    v2f A = {a0, a1};
    v2f Bv = { W[(long)ka * NOUT + col], W[(long)(ka + 1) * NOUT + col] };
    acc = __builtin_amdgcn_wmma_f32_16x16x4_f32(
        /*neg_a=*/false, A, /*neg_b=*/false, Bv,
        /*c_mod=*/(short)0, acc, /*reuse_a=*/false, /*reuse_b=*/false);
  }

  float bb = bias ? bias[col] : 0.0f;
  long rbase = (long)blockIdx.x * 16 + kh * 8;
  #pragma unroll
  for (int r = 0; r < 8; ++r)
    Y[(rbase + r) * NOUT + col] = acc[r] + bb;
}

// ---------------------------------------------------------------------------
// Helpers: fills, degree, rsqrt
// ---------------------------------------------------------------------------
__global__ void fill_f32(float* p, float v, long n) {
  long i = (long)blockIdx.x * blockDim.x + threadIdx.x;
  if (i < n) p[i] = v;
}
__global__ void fill_u8(unsigned char* p, unsigned char v, long n) {
  long i = (long)blockIdx.x * blockDim.x + threadIdx.x;
  if (i < n) p[i] = v;
}
__global__ void deg_edge(const int* __restrict__ dst,
                         const unsigned char* __restrict__ emask,
                         float* __restrict__ deg, long E) {
  long e = (long)blockIdx.x * blockDim.x + threadIdx.x;
  if (e < E && emask[e]) atomicAdd(&deg[dst[e]], 1.0f);
}
__global__ void rsqrt_k(float* p, long n) {
  long i = (long)blockIdx.x * blockDim.x + threadIdx.x;
  if (i < n) p[i] = rsqrtf(p[i]);
}

// OUT = 2*dinv^2 * H + b   (self-loop term, also initializes OUT for atomics)
__global__ void gcn_init(const float* __restrict__ H,
                         const float* __restrict__ dinv,
                         const float* __restrict__ b,
                         float* __restrict__ OUT, long total) {
  long i = (long)blockIdx.x * blockDim.x + threadIdx.x;
  if (i >= total) return;
  long nrow = i >> 7;
  int f = (int)(i & 127);
  float dv = dinv[nrow];
  OUT[i] = 2.0f * dv * dv * H[i] + b[f];
}

// wave-per-edge scatter: OUT[dst] += dinv[src]*dinv[dst] * H[src]
__global__ void gcn_edge(const float* __restrict__ H,
                         const float* __restrict__ dinv,
                         const int* __restrict__ src,
                         const int* __restrict__ dst,
                         const unsigned char* __restrict__ emask,
                         float* __restrict__ OUT, long E) {
  long gt = (long)blockIdx.x * blockDim.x + threadIdx.x;
  long e = gt >> 5;
  if (e >= E) return;
  if (!emask[e]) return;               // uniform across the wave
  int lane = threadIdx.x & 31;
  int s = src[e], d = dst[e];
  float nrm = dinv[s] * dinv[d];
  const float4* hs = (const float4*)(H + (long)s * HWF);
  float4 v = hs[lane];
  float* o = OUT + (long)d * HWF + lane * 4;
  atomicAdd(o + 0, nrm * v.x);
  atomicAdd(o + 1, nrm * v.y);
  atomicAdd(o + 2, nrm * v.z);
  atomicAdd(o + 3, nrm * v.w);
}

// wave-per-node: score = tanh(x . pw / ||pw||)
__global__ void score_kernel(const float* __restrict__ X,
                             const float* __restrict__ pw,
                             float* __restrict__ score, long n) {
  long gt = (long)blockIdx.x * blockDim.x + threadIdx.x;
  long node = gt >> 5;
  int lane = threadIdx.x & 31;
  if (node >= n) return;
  float s = 0.0f, w2 = 0.0f;
  for (int j = lane; j < HWF; j += 32) {
    float w = pw[j];
    s  += X[node * HWF + j] * w;
    w2 += w * w;
  }
  for (int off = 16; off > 0; off >>= 1) {
    s  += __shfl_down(s,  off, 32);
    w2 += __shfl_down(w2, off, 32);
  }
  if (lane == 0) score[node] = tanhf(s * rsqrtf(w2));
}

// one block per graph: bitonic sort scores descending, emit perm/skey/new_id
__global__ void topk_kernel(const float* __restrict__ score, int n_per, int k,
                            int* __restrict__ perm, float* __restrict__ skey,
                            int* __restrict__ new_id) {
  extern __shared__ char smem_raw[];
  float* key = (float*)smem_raw;
  int* idx = (int*)(smem_raw + (size_t)n_per * sizeof(float));
  int g = blockIdx.x;
  int t = threadIdx.x;
  for (int i = t; i < n_per; i += blockDim.x) {
    key[i] = score[(long)g * n_per + i];
    idx[i] = i;
    new_id[(long)g * n_per + i] = -1;
  }
  __syncthreads();
  for (int kk = 2; kk <= n_per; kk <<= 1) {
    for (int j = kk >> 1; j > 0; j >>= 1) {
      for (int i = t; i < n_per; i += blockDim.x) {
        int ixj = i ^ j;
        if (ixj > i) {
          bool up = (i & kk) == 0;       // descending overall
          float ki = key[i], kj = key[ixj];
          bool sw = up ? (ki < kj) : (ki > kj);
          if (sw) {
            key[i] = kj; key[ixj] = ki;
            int tm = idx[i]; idx[i] = idx[ixj]; idx[ixj] = tm;
          }
        }
      }
      __syncthreads();
    }
  }
  for (int jj = t; jj < k; jj += blockDim.x) {
    int old_local = idx[jj];
    perm[(long)g * k + jj] = g * n_per + old_local;
    skey[(long)g * k + jj] = key[jj];
    new_id[(long)g * n_per + old_local] = g * k + jj;
  }
}

// wave-per-new-node gather: Y[new] = X[perm[new]] * skey[new]
__global__ void gather_kernel(const float* __restrict__ X,
                              const int* __restrict__ perm,
                              const float* __restrict__ skey,
                              float* __restrict__ Y, long n_new) {
  long gt = (long)blockIdx.x * blockDim.x + threadIdx.x;
  long node = gt >> 5;
  int lane = threadIdx.x & 31;
  if (node >= n_new) return;
  int o = perm[node];
  float s = skey[node];
  const float4* xs = (const float4*)(X + (long)o * HWF);
  float4* ys = (float4*)(Y + node * HWF);
  float4 v = xs[lane];
  float4 r; r.x = v.x * s; r.y = v.y * s; r.z = v.z * s; r.w = v.w * s;
  ys[lane] = r;
}

__global__ void remap_kernel(int* __restrict__ src, int* __restrict__ dst,
                             unsigned char* __restrict__ emask,
                             const int* __restrict__ new_id, long E) {
  long e = (long)blockIdx.x * blockDim.x + threadIdx.x;
  if (e >= E) return;
  int ns = new_id[src[e]];
  int nd = new_id[dst[e]];
  bool valid = (ns >= 0) && (nd >= 0) && (emask[e] != 0);
  emask[e] = valid ? 1 : 0;
  src[e] = valid ? ns : 0;
  dst[e] = valid ? nd : 0;
}

// one block (128 threads) per graph: mean-pool -> Linear -> BN -> ReLU -> Linear
__global__ void readout_kernel(const float* __restrict__ X, int kpg,
                               const float* __restrict__ W1, const float* __restrict__ b1,
                               const float* __restrict__ gamma, const float* __restrict__ beta,
                               const float* __restrict__ mean, const float* __restrict__ var,
                               const float* __restrict__ W2, const float* __restrict__ b2,
                               float* __restrict__ out) {
  __shared__ float r[HWF];
  __shared__ float h1[64];
  int g = blockIdx.x, t = threadIdx.x;
  float s = 0.0f;
  for (int j = 0; j < kpg; ++j) s += X[((long)g * kpg + j) * HWF + t];
  r[t] = s / (float)kpg;
  __syncthreads();
  if (t < 64) {
    float a = b1[t];
    for (int kk = 0; kk < HWF; ++kk) a += r[kk] * W1[kk * 64 + t];
    float sc = gamma[t] * rsqrtf(var[t] + EPSV);
    a = (a - mean[t]) * sc + beta[t];
    h1[t] = fmaxf(a, 0.0f);
  }
  __syncthreads();
  if (t < 4) {
    float a = b2[t];
    for (int kk = 0; kk < 64; ++kk) a += h1[kk] * W2[kk * 4 + t];
    out[g * 4 + t] = a;
  }
}

// ---------------------------------------------------------------------------
static inline size_t align256(size_t x) { return (x + 255) & ~(size_t)255; }

extern "C" void kernel_launch(void* const* d_in, const int* in_sizes, int n_in,
                              void* d_out, int out_size, void* d_ws, size_t ws_size,
                              hipStream_t stream) {
  (void)n_in; (void)out_size; (void)ws_size;
  const float* x_in     = (const float*)d_in[0];
  const int*   edge_idx = (const int*)d_in[1];
  const float* t0_gamma = (const float*)d_in[3];
  const float* t0_beta  = (const float*)d_in[4];
  const float* t0_mean  = (const float*)d_in[5];
  const float* t0_var   = (const float*)d_in[6];
  const float* t0_W     = (const float*)d_in[7];
  const float* t0_b     = (const float*)d_in[8];
  const float* t_gamma  = (const float*)d_in[9];
  const float* t_beta   = (const float*)d_in[10];
  const float* t_mean   = (const float*)d_in[11];
  const float* t_var    = (const float*)d_in[12];
  const float* t_W      = (const float*)d_in[13];
  const float* t_b      = (const float*)d_in[14];
  const float* conv_W   = (const float*)d_in[15];
  const float* conv_b   = (const float*)d_in[16];
  const float* pool_w   = (const float*)d_in[17];
  const float* p_W1     = (const float*)d_in[18];
  const float* p_b1     = (const float*)d_in[19];
  const float* p_gamma  = (const float*)d_in[20];
  const float* p_beta   = (const float*)d_in[21];
  const float* p_mean   = (const float*)d_in[22];
  const float* p_var    = (const float*)d_in[23];
  const float* p_W2     = (const float*)d_in[24];
  const float* p_b2     = (const float*)d_in[25];

  const int Bg = 64, N0 = 2048, IW = 64;
  const long n0 = (long)Bg * N0;            // 131072
  const long E  = (long)in_sizes[1] / 2;    // 2097152

  // workspace partition
  char* w = (char*)d_ws;
  size_t off = 0;
  float* B1 = (float*)(w + off); off = align256(off + (size_t)n0 * HWF * 4);
  float* B2 = (float*)(w + off); off = align256(off + (size_t)n0 * HWF * 4);
  float* B3 = (float*)(w + off); off = align256(off + (size_t)n0 * HWF * 4);
  float* dinv  = (float*)(w + off); off = align256(off + (size_t)n0 * 4);
  float* score = (float*)(w + off); off = align256(off + (size_t)n0 * 4);
  int*   newid = (int*)(w + off);   off = align256(off + (size_t)n0 * 4);
  int*   perm  = (int*)(w + off);   off = align256(off + (size_t)(n0 / 2) * 4);
  float* skey  = (float*)(w + off); off = align256(off + (size_t)(n0 / 2) * 4);
  int*   srcw  = (int*)(w + off);   off = align256(off + (size_t)E * 4);
  int*   dstw  = (int*)(w + off);   off = align256(off + (size_t)E * 4);
  unsigned char* emask = (unsigned char*)(w + off); off = align256(off + (size_t)E);

  hipMemcpyAsync(srcw, edge_idx,     E * sizeof(int), hipMemcpyDeviceToDevice, stream);
  hipMemcpyAsync(dstw, edge_idx + E, E * sizeof(int), hipMemcpyDeviceToDevice, stream);
  fill_u8<<<(int)((E + 255) / 256), 256, 0, stream>>>(emask, 1, E);

  const float* xcur = x_in;
  long n = n0;
  int n_per = N0;

  for (int d = 0; d < 4; ++d) {
    // symmetric-normalization degree (shared by both GCNs of this depth)
    fill_f32<<<(int)((n + 255) / 256), 256, 0, stream>>>(dinv, 2.0f, n);
    deg_edge<<<(int)((E + 255) / 256), 256, 0, stream>>>(dstw, emask, dinv, E);
    rsqrt_k<<<(int)((n + 255) / 256), 256, 0, stream>>>(dinv, n);

    for (int half = 0; half < 2; ++half) {
      int j = 2 * d + half;
      const float *g_, *bt_, *mn_, *vr_, *Wt_, *bb_;
      int K;
      if (j == 0) {
        g_ = t0_gamma; bt_ = t0_beta; mn_ = t0_mean; vr_ = t0_var;
        Wt_ = t0_W; bb_ = t0_b; K = IW;
      } else {
        g_  = t_gamma + (long)(j - 1) * HWF;
        bt_ = t_beta  + (long)(j - 1) * HWF;
        mn_ = t_mean  + (long)(j - 1) * HWF;
        vr_ = t_var   + (long)(j - 1) * HWF;
        Wt_ = t_W     + (long)(j - 1) * HWF * HWF;
        bb_ = t_b     + (long)(j - 1) * HWF;
        K = HWF;
      }
      // BN -> ReLU -> Linear  (WMMA fp32)
      lin16_wmma<<<(int)(n / 16), 256, 0, stream>>>(
          xcur, K, Wt_, HWF, g_, bt_, mn_, vr_, bb_, B1, 1);
      // GCN: h = x @ conv_W  (WMMA fp32, no activation/bias)
      lin16_wmma<<<(int)(n / 16), 256, 0, stream>>>(
          B1, HWF, conv_W + (long)j * HWF * HWF, HWF,
          nullptr, nullptr, nullptr, nullptr, nullptr, B2, 0);
      // self-loop init + bias, then edge scatter with atomics
      gcn_init<<<(int)((n * HWF + 255) / 256), 256, 0, stream>>>(
          B2, dinv, conv_b + (long)j * HWF, B3, n * HWF);
      gcn_edge<<<(int)((E * 32 + 255) / 256), 256, 0, stream>>>(
          B2, dinv, srcw, dstw, emask, B3, E);
      xcur = B3;
    }

    // TopK pool (ratio 0.5)
    int kkeep = n_per / 2;
    score_kernel<<<(int)((n * 32 + 255) / 256), 256, 0, stream>>>(
        B3, pool_w + (long)d * HWF, score, n);
    topk_kernel<<<Bg, 256, (size_t)n_per * 8, stream>>>(
        score, n_per, kkeep, perm, skey, newid);
    gather_kernel<<<(int)(((n / 2) * 32 + 255) / 256), 256, 0, stream>>>(
        B3, perm, skey, B2, n / 2);
    remap_kernel<<<(int)((E + 255) / 256), 256, 0, stream>>>(
        srcw, dstw, emask, newid, E);

    n /= 2;
    n_per /= 2;
    xcur = B2;            // next depth reads pooled x from B2
  }

  // readout: mean-pool per graph + predictor MLP
  readout_kernel<<<Bg, HWF, 0, stream>>>(
      B2, (int)(n / Bg), p_W1, p_b1, p_gamma, p_beta, p_mean, p_var,
      p_W2, p_b2, (float*)d_out);
}